// AdaptiveHeatmapLossFromCenters_87978110091318
// MI455X (gfx1250) — compile-verified
//
#include <hip/hip_runtime.h>
#include <math.h>

#define REF_GROUND_RES 0.2f
#define PR_MIN 0.2f

typedef __attribute__((ext_vector_type(2))) float v2f;
typedef __attribute__((ext_vector_type(8))) float v8f;

// ---------------------------------------------------------------------------
// Zero the two scalar loss accumulators (d_out is poisoned by the harness).
// ---------------------------------------------------------------------------
__global__ void init_losses_kernel(float* out) {
    if (threadIdx.x < 2) out[threadIdx.x] = 0.0f;
}

// ---------------------------------------------------------------------------
// Per-center prep: gather sm at clamped center coords, compute sigma and the
// rank-4 "A" features of the fused score GEMM:
//   score[m,n] = -s*(y^2+x^2) + 2*s*cy*y + 2*s*cx*x - s*(cy^2+cx^2)
//              = -d2 / (2*sigma^2)
// A[m] = { -s, 2*s*cy, 2*s*cx, -s*(cy^2+cx^2) }   (one float4 per center)
// ---------------------------------------------------------------------------
__global__ void prep_centers_kernel(const float* __restrict__ sm,
                                    const float* __restrict__ gr,
                                    const int*   __restrict__ centers,
                                    float4*      __restrict__ params,
                                    int B, int H, int W, int N) {
    int tid = blockIdx.x * blockDim.x + threadIdx.x;
    if (tid >= B * N) return;
    int b = tid / N;
    int n = tid - b * N;
    int cy = centers[(b * N + n) * 2 + 0];
    int cx = centers[(b * N + n) * 2 + 1];
    cy = min(max(cy, 0), H - 1);      // reference clamps both coords with H-1
    cx = min(max(cx, 0), H - 1);
    float smv   = sm[(long)b * H * W + (long)cy * W + cx];
    float g     = gr[b];
    float sigma = PR_MIN / g + fmaxf(smv, 0.0f) * (REF_GROUND_RES / g);
    float s     = 1.0f / (2.0f * sigma * sigma);
    float fy = (float)cy, fx = (float)cx;
    params[tid] = make_float4(-s, 2.0f * s * fy, 2.0f * s * fx,
                              -s * (fy * fy + fx * fx));
}

// ---------------------------------------------------------------------------
// Wave/block loss reduction + atomics (shared by both kernels).
// ---------------------------------------------------------------------------
__device__ __forceinline__
void reduce_losses(float sl, float hl, float* losses, float inv_bhw,
                   int lane, int waveInBlock) {
    #pragma unroll
    for (int off = 16; off >= 1; off >>= 1) {
        hl += __shfl_xor(hl, off, 32);
        sl += __shfl_xor(sl, off, 32);
    }
    __shared__ float red[2][8];
    if (lane == 0) { red[0][waveInBlock] = sl; red[1][waveInBlock] = hl; }
    __syncthreads();
    if (threadIdx.x == 0) {
        float tsl = 0.0f, thl = 0.0f;
        const int nw = blockDim.x >> 5;
        for (int i = 0; i < nw; ++i) { tsl += red[0][i]; thl += red[1][i]; }
        atomicAdd(&losses[0], tsl * inv_bhw);
        atomicAdd(&losses[1], thl * inv_bhw);
    }
}

// ---------------------------------------------------------------------------
// Specialized kernel: NT center tiles (N = NT*16), T pixel tiles per wave.
// The NT "A" operands (2 VGPRs each) are loaded ONCE per wave and reused
// across T pixel tiles -> 8x less param cache traffic, WMMA loop is
// register-only: build B operand, NT back-to-back v_wmma_f32_16x16x4_f32,
// v_max3 reduction, shfl_xor(16), expf, coalesced stores.
// ---------------------------------------------------------------------------
template <int NT, int T>
__global__ void __launch_bounds__(256)
heatmap_loss_tiled(const float*  __restrict__ hm,
                   const float*  __restrict__ sm,
                   const float*  __restrict__ mask,
                   const float4* __restrict__ params,
                   float*        __restrict__ losses,   // [0]=scale, [1]=hm
                   float*        __restrict__ out_gt,
                   int B, int H, int W, float inv_bhw) {
    const int HW          = H * W;
    const int grpPerSamp  = HW / (16 * T);
    const int lane        = threadIdx.x & 31;
    const int waveInBlock = threadIdx.x >> 5;
    const int grp         = blockIdx.x * (blockDim.x >> 5) + waveInBlock;

    float sl = 0.0f, hl = 0.0f;

    if (grp < B * grpPerSamp) {         // wave-uniform: EXEC stays all-1s
        const int  b      = grp / grpPerSamp;
        const int  pbase0 = (grp - b * grpPerSamp) * (16 * T);
        const int  col    = lane & 15;
        const bool hiK    = (lane >= 16);

        // ---- preload all A operands (centers) into registers -------------
        v2f am[NT];
        const float4* pp = params + (long)b * (NT * 16);
        #pragma unroll
        for (int i = 0; i < NT; ++i) {
            float4 f = pp[i * 16 + col];
            am[i][0] = hiK ? f.z : f.x;     // K=2 : K=0
            am[i][1] = hiK ? f.w : f.y;     // K=3 : K=1
        }

        // ---- sweep T pixel tiles with A resident in VGPRs ----------------
        #pragma unroll
        for (int t = 0; t < T; ++t) {
            const int   p   = pbase0 + t * 16 + col;
            const int   row = p / W;
            const float y   = (float)row;
            const float x   = (float)(p - row * W);

            v2f bm;                               // B operand (pixels)
            bm[0] = hiK ? x    : (y * y + x * x); // K=2 : K=0
            bm[1] = hiK ? 1.0f : y;               // K=3 : K=1

            v8f cmax;
            #pragma unroll
            for (int i = 0; i < 8; ++i) cmax[i] = -3.402823466e38f;

            #pragma unroll
            for (int i = 0; i < NT; ++i) {
                v8f cz = {};
                v8f d = __builtin_amdgcn_wmma_f32_16x16x4_f32(
                            false, am[i], false, bm, (short)0, cz,
                            false, false);
                #pragma unroll
                for (int k = 0; k < 8; ++k) cmax[k] = fmaxf(cmax[k], d[k]);
            }

            float m = cmax[0];
            #pragma unroll
            for (int i = 1; i < 8; ++i) m = fmaxf(m, cmax[i]);
            m = fmaxf(m, __shfl_xor(m, 16, 32));

            const float gt   = expf(m);
            const long  gidx = (long)b * HW + p;
            if (!hiK) {                      // lanes 0-15: gt store + hm loss
                out_gt[gidx] = gt;
                float dv = hm[gidx] - gt;
                hl += dv * dv * mask[gidx];
            } else {                         // lanes 16-31: scale loss
                float sv = sm[gidx];
                sl += sv * sv;
            }
        }
    }
    reduce_losses(sl, hl, losses, inv_bhw, lane, waveInBlock);
}

// ---------------------------------------------------------------------------
// Generic fallback: one 16-pixel tile per wave, dynamic N (multiple of 16).
// ---------------------------------------------------------------------------
__global__ void __launch_bounds__(256)
heatmap_loss_generic(const float*  __restrict__ hm,
                     const float*  __restrict__ sm,
                     const float*  __restrict__ mask,
                     const float4* __restrict__ params,
                     float*        __restrict__ losses,
                     float*        __restrict__ out_gt,
                     int B, int H, int W, int N, float inv_bhw) {
    const int HW           = H * W;
    const int tilesPerSamp = HW >> 4;
    const int lane         = threadIdx.x & 31;
    const int waveInBlock  = threadIdx.x >> 5;
    const int wave         = blockIdx.x * (blockDim.x >> 5) + waveInBlock;

    float sl = 0.0f, hl = 0.0f;

    if (wave < B * tilesPerSamp) {
        const int  b     = wave / tilesPerSamp;
        const int  pbase = (wave - b * tilesPerSamp) << 4;
        const int  col   = lane & 15;
        const bool hiK   = (lane >= 16);
        const int  p     = pbase + col;
        const int  row   = p / W;
        const float y    = (float)row;
        const float x    = (float)(p - row * W);

        v2f bm;
        bm[0] = hiK ? x    : (y * y + x * x);
        bm[1] = hiK ? 1.0f : y;

        v8f cmax;
        #pragma unroll
        for (int i = 0; i < 8; ++i) cmax[i] = -3.402823466e38f;

        const float4* pp = params + (long)b * N;
        for (int ct = 0; ct < N; ct += 16) {
            float4 f = pp[ct + col];
            v2f am;
            am[0] = hiK ? f.z : f.x;
            am[1] = hiK ? f.w : f.y;
            v8f cz = {};
            v8f d = __builtin_amdgcn_wmma_f32_16x16x4_f32(
                        false, am, false, bm, (short)0, cz, false, false);
            #pragma unroll
            for (int i = 0; i < 8; ++i) cmax[i] = fmaxf(cmax[i], d[i]);
        }

        float m = cmax[0];
        #pragma unroll
        for (int i = 1; i < 8; ++i) m = fmaxf(m, cmax[i]);
        m = fmaxf(m, __shfl_xor(m, 16, 32));

        const float gt   = expf(m);
        const long  gidx = (long)b * HW + p;
        if (!hiK) {
            out_gt[gidx] = gt;
            float dv = hm[gidx] - gt;
            hl = dv * dv * mask[gidx];
        } else {
            float sv = sm[gidx];
            sl = sv * sv;
        }
    }
    reduce_losses(sl, hl, losses, inv_bhw, lane, waveInBlock);
}

// ---------------------------------------------------------------------------
// Host launcher
// ---------------------------------------------------------------------------
extern "C" void kernel_launch(void* const* d_in, const int* in_sizes, int n_in,
                              void* d_out, int out_size, void* d_ws, size_t ws_size,
                              hipStream_t stream) {
    const float* hm      = (const float*)d_in[0];   // pred_hm   [B,1,H,W]
    const float* sm      = (const float*)d_in[1];   // pred_sm   [B,1,H,W]
    const float* gr      = (const float*)d_in[2];   // ground_resolution [B]
    const float* mask    = (const float*)d_in[3];   // mask      [B,1,H,W]
    const int*   centers = (const int*)d_in[4];     // centers   [B,N,2]

    const int  B  = in_sizes[2];
    const long HW = (long)in_sizes[0] / B;
    const int  H  = (int)(sqrt((double)HW) + 0.5);  // square images
    const int  W  = H;
    const int  N  = in_sizes[4] / (B * 2);

    float*  out    = (float*)d_out;
    float*  out_gt = out + 2;                       // [mean_sl, mean_hl, gts...]
    float4* params = (float4*)d_ws;                 // B*N float4 (16 KB)

    init_losses_kernel<<<1, 32, 0, stream>>>(out);
    prep_centers_kernel<<<(B * N + 255) / 256, 256, 0, stream>>>(
        sm, gr, centers, params, B, H, W, N);

    const float inv_bhw       = 1.0f / (float)((long)B * H * W);
    const int   wavesPerBlock = 8;                  // 256 threads

    constexpr int NT = 8;                           // 128 centers
    constexpr int T  = 8;                           // 128 pixels per wave
    if (N == NT * 16 && (HW % (16 * T)) == 0 && (W % (16 * T)) == 0) {
        const int groups = B * (int)(HW / (16 * T));
        const int blocks = (groups + wavesPerBlock - 1) / wavesPerBlock;
        heatmap_loss_tiled<NT, T><<<blocks, 256, 0, stream>>>(
            hm, sm, mask, params, out, out_gt, B, H, W, inv_bhw);
    } else {
        const int ntiles = B * (int)(HW / 16);
        const int blocks = (ntiles + wavesPerBlock - 1) / wavesPerBlock;
        heatmap_loss_generic<<<blocks, 256, 0, stream>>>(
            hm, sm, mask, params, out, out_gt, B, H, W, N, inv_bhw);
    }
}